// TGPYRegressor_68332929679711
// MI455X (gfx1250) — compile-verified
//
#include <hip/hip_runtime.h>
#include <math.h>

// Problem constants (fixed by setup_inputs)
#define Nn 1024   // train points
#define Mm 2048   // test points
#define Dd 4      // feature dim
#define Cc 64     // chains
#define NBk 64    // Cholesky / TRSM block size

typedef __attribute__((ext_vector_type(2))) float v2f;
typedef __attribute__((ext_vector_type(8))) float v8f;

// ---------------------------------------------------------------------------
// Init accumulators
// ---------------------------------------------------------------------------
__global__ void zero_acc_kernel(float* accM, float* accV) {
  int j = blockIdx.x * blockDim.x + threadIdx.x;
  if (j < Mm) { accM[j] = 0.0f; accV[j] = 0.0f; }
}

// ---------------------------------------------------------------------------
// K[i,j] = var*exp(-0.5*||xi-xj||^2/ls^2) + noise*(i==j)
// ---------------------------------------------------------------------------
__global__ void build_K_kernel(const float* __restrict__ Xtr,
                               const float* __restrict__ ls,
                               const float* __restrict__ var,
                               const float* __restrict__ noi,
                               int c, float* __restrict__ K) {
  int idx = blockIdx.x * 256 + threadIdx.x;
  int i = idx >> 10;           // / Nn
  int j = idx & (Nn - 1);
  float l = ls[c];
  float inv = 0.5f / (l * l);
  float d2 = 0.0f;
#pragma unroll
  for (int k = 0; k < Dd; ++k) {
    float t = Xtr[i * Dd + k] - Xtr[j * Dd + k];
    d2 += t * t;
  }
  float v = var[c] * __expf(-d2 * inv);
  if (i == j) v += noi[c];
  K[i * Nn + j] = v;
}

// ---------------------------------------------------------------------------
// B[i,j] = var*exp(-0.5*||xtr_i - xte_j||^2/ls^2)   (n x m, row-major ld=Mm)
// ---------------------------------------------------------------------------
__global__ void build_B_kernel(const float* __restrict__ Xtr,
                               const float* __restrict__ Xte,
                               const float* __restrict__ ls,
                               const float* __restrict__ var,
                               int c, float* __restrict__ B) {
  int idx = blockIdx.x * 256 + threadIdx.x;
  int i = idx >> 11;           // / Mm
  int j = idx & (Mm - 1);
  float l = ls[c];
  float inv = 0.5f / (l * l);
  float d2 = 0.0f;
#pragma unroll
  for (int k = 0; k < Dd; ++k) {
    float t = Xtr[i * Dd + k] - Xte[j * Dd + k];
    d2 += t * t;
  }
  B[i * Mm + j] = var[c] * __expf(-d2 * inv);
}

// ---------------------------------------------------------------------------
// Cholesky: factor 64x64 diagonal block in LDS (one WG of 64 threads)
// ---------------------------------------------------------------------------
__global__ void chol_diag_kernel(float* __restrict__ K, int k0) {
  __shared__ float A[NBk][NBk + 1];
  int tid = threadIdx.x;           // owns row tid
  for (int j = 0; j < NBk; ++j) A[tid][j] = K[(k0 + tid) * Nn + k0 + j];
  __syncthreads();
  for (int j = 0; j < NBk; ++j) {
    if (tid == j) A[j][j] = sqrtf(A[j][j]);
    __syncthreads();
    if (tid > j) A[tid][j] /= A[j][j];
    __syncthreads();
    if (tid > j) {
      float f = A[tid][j];
      for (int p = j + 1; p <= tid; ++p) A[tid][p] -= f * A[p][j];
    }
    __syncthreads();
  }
  for (int j = 0; j <= tid; ++j) K[(k0 + tid) * Nn + k0 + j] = A[tid][j];
}

// ---------------------------------------------------------------------------
// Cholesky panel: L21 = A21 * L11^-T  (thread per row, row cached in LDS)
// ---------------------------------------------------------------------------
__global__ void chol_panel_kernel(float* __restrict__ K, int k0) {
  __shared__ float L11[NBk][NBk + 1];
  __shared__ float R[NBk][NBk + 1];
  int tid = threadIdx.x;
  int row = k0 + NBk + blockIdx.x * NBk + tid;
  for (int j = 0; j < NBk; ++j) L11[tid][j] = K[(k0 + tid) * Nn + k0 + j];
  for (int j = 0; j < NBk; ++j) R[tid][j] = K[row * Nn + k0 + j];
  __syncthreads();
  for (int j = 0; j < NBk; ++j) {
    float s = R[tid][j];
    for (int p = 0; p < j; ++p) s -= R[tid][p] * L11[j][p];
    R[tid][j] = s / L11[j][j];
  }
  for (int j = 0; j < NBk; ++j) K[row * Nn + k0 + j] = R[tid][j];
}

// ---------------------------------------------------------------------------
// SYRK trailing update: A22 -= L21 * L21^T  (block-lower 32x32 macro-tiles)
// Each wave computes a 32x32 tile = 4 f32 WMMA accumulators; 2 A-frags and
// 2 B-frags per K-step are each reused twice (halves vmem per WMMA).
// Subtraction realized by negating the A fragment (f32 WMMA has C-neg only).
// ---------------------------------------------------------------------------
__global__ void syrk_update_kernel(float* __restrict__ K, int k0) {
  int t0 = k0 + NBk;
  int t32 = (Nn - t0) >> 5;
  int wid = threadIdx.x >> 5;
  int lane = threadIdx.x & 31;
  int i32 = blockIdx.y * 4 + wid;   // macro-tile row
  int j32 = blockIdx.x;             // macro-tile col
  if (i32 >= t32 || i32 < j32) return;
  int row0 = t0 + i32 * 32;
  int col0 = t0 + j32 * 32;
  int lh = lane & 15;
  int hi = lane >> 4;               // 0: K pair {0,1}, 1: K pair {2,3}
  // C tile is read-modify-written after the K loop: prefetch its 32 rows
  __builtin_prefetch(&K[(row0 + lane) * Nn + col0], 1, 3);
  v8f acc00 = {}, acc01 = {}, acc10 = {}, acc11 = {};
  for (int kk = 0; kk < NBk; kk += 4) {
    int ka = k0 + kk + 2 * hi;
    v2f a0, a1, b0, b1;
    // A fragments (16x4): M striped over lanes 0-15, K split by lane half
    a0.x = -K[(row0 + lh) * Nn + ka];
    a0.y = -K[(row0 + lh) * Nn + ka + 1];
    a1.x = -K[(row0 + 16 + lh) * Nn + ka];
    a1.y = -K[(row0 + 16 + lh) * Nn + ka + 1];
    // B fragments (4x16) = L21^T : B[k][n] = L[col0+n][k0+k]
    b0.x = K[(col0 + lh) * Nn + ka];
    b0.y = K[(col0 + lh) * Nn + ka + 1];
    b1.x = K[(col0 + 16 + lh) * Nn + ka];
    b1.y = K[(col0 + 16 + lh) * Nn + ka + 1];
    acc00 = __builtin_amdgcn_wmma_f32_16x16x4_f32(false, a0, false, b0,
                                                  (short)0, acc00, false, false);
    acc01 = __builtin_amdgcn_wmma_f32_16x16x4_f32(false, a0, false, b1,
                                                  (short)0, acc01, false, false);
    acc10 = __builtin_amdgcn_wmma_f32_16x16x4_f32(false, a1, false, b0,
                                                  (short)0, acc10, false, false);
    acc11 = __builtin_amdgcn_wmma_f32_16x16x4_f32(false, a1, false, b1,
                                                  (short)0, acc11, false, false);
  }
#pragma unroll
  for (int r = 0; r < 8; ++r) {
    int ra = row0 + hi * 8 + r;
    int rb = row0 + 16 + hi * 8 + r;
    K[ra * Nn + col0 + lh]      += acc00[r];
    K[ra * Nn + col0 + 16 + lh] += acc01[r];
    K[rb * Nn + col0 + lh]      += acc10[r];
    K[rb * Nn + col0 + 16 + lh] += acc11[r];
  }
}

// ---------------------------------------------------------------------------
// alpha = L^-T (L^-1 y)   — one 1024-thread workgroup, split barriers
// ---------------------------------------------------------------------------
__global__ void solve_alpha_kernel(const float* __restrict__ K,
                                   const float* __restrict__ y,
                                   float* __restrict__ alpha) {
  __shared__ float xs[Nn];
  int i = threadIdx.x;
  float s = 0.0f;
  for (int j = 0; j < Nn; ++j) {            // forward: L z = y
    if (i == j) xs[j] = (y[j] - s) / K[j * Nn + j];
    __syncthreads();
    if (i > j) s += K[i * Nn + j] * xs[j];
  }
  __syncthreads();
  s = 0.0f;
  for (int jj = 0; jj < Nn; ++jj) {         // backward: L^T a = z (in place)
    int j = Nn - 1 - jj;
    if (i == j) xs[j] = (xs[j] - s) / K[j * Nn + j];
    __syncthreads();
    if (i < j) s += K[j * Nn + i] * xs[j];  // L^T[i][j] = L[j][i] (coalesced)
  }
  __syncthreads();
  alpha[i] = xs[i];
}

// ---------------------------------------------------------------------------
// accM[j] += sum_i B[i,j] * alpha[i]
// ---------------------------------------------------------------------------
__global__ void mean_acc_kernel(const float* __restrict__ B,
                                const float* __restrict__ alpha,
                                float* __restrict__ accM) {
  int j = blockIdx.x * 256 + threadIdx.x;
  float s = 0.0f;
  for (int i = 0; i < Nn; ++i) s += B[i * Mm + j] * alpha[i];
  accM[j] += s;
}

// ---------------------------------------------------------------------------
// TRSM diagonal block: V[k0..k0+63, :] = L11^-1 * B[k0..k0+63, :]
// thread-per-column, column kept in LDS (banked by tid)
// ---------------------------------------------------------------------------
__global__ void trsm_diag_kernel(const float* __restrict__ K,
                                 float* __restrict__ B, int k0) {
  __shared__ float L11[NBk][NBk + 1];
  __shared__ float col[NBk][NBk + 1];
  int tid = threadIdx.x;
  int j = blockIdx.x * NBk + tid;
  for (int p = 0; p < NBk; ++p) L11[tid][p] = K[(k0 + tid) * Nn + k0 + p];
  for (int r = 0; r < NBk; ++r) col[r][tid] = B[(k0 + r) * Mm + j];
  __syncthreads();
  for (int r = 0; r < NBk; ++r) {
    float s = col[r][tid];
    for (int p = 0; p < r; ++p) s -= L11[r][p] * col[p][tid];
    col[r][tid] = s / L11[r][r];
  }
  for (int r = 0; r < NBk; ++r) B[(k0 + r) * Mm + j] = col[r][tid];
}

// ---------------------------------------------------------------------------
// TRSM trailing update: B[t0.., :] -= L[t0.., k0..k0+63] * V[k0..k0+63, :]
// WMMA GEMM, 32x32 macro-tile per wave (4 accumulators, 2x operand reuse)
// ---------------------------------------------------------------------------
__global__ void trsm_update_kernel(const float* __restrict__ K,
                                   float* __restrict__ B, int k0) {
  int t0 = k0 + NBk;
  int mt32 = (Nn - t0) >> 5;
  int wid = threadIdx.x >> 5;
  int lane = threadIdx.x & 31;
  int i32 = blockIdx.y * 4 + wid;
  int j32 = blockIdx.x;
  if (i32 >= mt32) return;
  int row0 = t0 + i32 * 32;
  int col0 = j32 * 32;
  int lh = lane & 15;
  int hi = lane >> 4;
  // C tile is read-modify-written after the K loop: prefetch its 32 rows
  __builtin_prefetch(&B[(row0 + lane) * Mm + col0], 1, 3);
  v8f acc00 = {}, acc01 = {}, acc10 = {}, acc11 = {};
  for (int kk = 0; kk < NBk; kk += 4) {
    int ka = k0 + kk + 2 * hi;
    v2f a0, a1, b0, b1;
    a0.x = -K[(row0 + lh) * Nn + ka];        // negated A -> C -= A*B
    a0.y = -K[(row0 + lh) * Nn + ka + 1];
    a1.x = -K[(row0 + 16 + lh) * Nn + ka];
    a1.y = -K[(row0 + 16 + lh) * Nn + ka + 1];
    b0.x = B[ka * Mm + col0 + lh];
    b0.y = B[(ka + 1) * Mm + col0 + lh];
    b1.x = B[ka * Mm + col0 + 16 + lh];
    b1.y = B[(ka + 1) * Mm + col0 + 16 + lh];
    acc00 = __builtin_amdgcn_wmma_f32_16x16x4_f32(false, a0, false, b0,
                                                  (short)0, acc00, false, false);
    acc01 = __builtin_amdgcn_wmma_f32_16x16x4_f32(false, a0, false, b1,
                                                  (short)0, acc01, false, false);
    acc10 = __builtin_amdgcn_wmma_f32_16x16x4_f32(false, a1, false, b0,
                                                  (short)0, acc10, false, false);
    acc11 = __builtin_amdgcn_wmma_f32_16x16x4_f32(false, a1, false, b1,
                                                  (short)0, acc11, false, false);
  }
#pragma unroll
  for (int r = 0; r < 8; ++r) {
    int ra = row0 + hi * 8 + r;
    int rb = row0 + 16 + hi * 8 + r;
    B[ra * Mm + col0 + lh]      += acc00[r];
    B[ra * Mm + col0 + 16 + lh] += acc01[r];
    B[rb * Mm + col0 + lh]      += acc10[r];
    B[rb * Mm + col0 + 16 + lh] += acc11[r];
  }
}

// ---------------------------------------------------------------------------
// accV[j] += sum_i V[i,j]^2
// ---------------------------------------------------------------------------
__global__ void var_acc_kernel(const float* __restrict__ B,
                               float* __restrict__ accV) {
  int j = blockIdx.x * 256 + threadIdx.x;
  float s = 0.0f;
  for (int i = 0; i < Nn; ++i) { float v = B[i * Mm + j]; s += v * v; }
  accV[j] += s;
}

// ---------------------------------------------------------------------------
// out = [meanAcc/C , varAcc/C]
// ---------------------------------------------------------------------------
__global__ void finalize_kernel(const float* __restrict__ accM,
                                const float* __restrict__ accV,
                                float* __restrict__ out) {
  int j = blockIdx.x * 256 + threadIdx.x;
  if (j < Mm) {
    const float inv = 1.0f / (float)Cc;
    out[j] = accM[j] * inv;
    out[Mm + j] = accV[j] * inv;
  }
}

// ---------------------------------------------------------------------------
extern "C" void kernel_launch(void* const* d_in, const int* in_sizes, int n_in,
                              void* d_out, int out_size, void* d_ws, size_t ws_size,
                              hipStream_t stream) {
  const float* Xtr = (const float*)d_in[0];   // (1024,4)
  const float* ytr = (const float*)d_in[1];   // (1024,)
  const float* Xte = (const float*)d_in[2];   // (2048,4)
  const float* ls  = (const float*)d_in[3];   // (64,)
  const float* var = (const float*)d_in[4];   // (64,)
  const float* noi = (const float*)d_in[5];   // (64,)
  float* out = (float*)d_out;                  // 2048 means ++ 2048 vars

  // Workspace layout (floats): K | B | alpha | accM | accV  (~12.6 MB)
  float* K     = (float*)d_ws;
  float* B     = K + (size_t)Nn * Nn;
  float* alpha = B + (size_t)Nn * Mm;
  float* accM  = alpha + Nn;
  float* accV  = accM + Mm;

  zero_acc_kernel<<<(Mm + 255) / 256, 256, 0, stream>>>(accM, accV);

  for (int c = 0; c < Cc; ++c) {
    // 1) kernel matrix with noise
    build_K_kernel<<<(Nn * Nn) / 256, 256, 0, stream>>>(Xtr, ls, var, noi, c, K);

    // 2) blocked Cholesky (right-looking), WMMA SYRK trailing update
    for (int k0 = 0; k0 < Nn; k0 += NBk) {
      chol_diag_kernel<<<1, NBk, 0, stream>>>(K, k0);
      if (k0 + NBk < Nn) {
        int T = Nn - k0 - NBk;
        chol_panel_kernel<<<T / NBk, NBk, 0, stream>>>(K, k0);
        int t32 = T >> 5;
        dim3 g(t32, (t32 + 3) / 4);
        syrk_update_kernel<<<g, 128, 0, stream>>>(K, k0);
      }
    }

    // 3) alpha = K^-1 y
    solve_alpha_kernel<<<1, Nn, 0, stream>>>(K, ytr, alpha);

    // 4) cross kernel + predictive mean accumulation
    build_B_kernel<<<(Nn * Mm) / 256, 256, 0, stream>>>(Xtr, Xte, ls, var, c, B);
    mean_acc_kernel<<<Mm / 256, 256, 0, stream>>>(B, alpha, accM);

    // 5) V = L^-1 B (blocked forward TRSM, WMMA update)
    for (int k0 = 0; k0 < Nn; k0 += NBk) {
      trsm_diag_kernel<<<Mm / NBk, NBk, 0, stream>>>(K, B, k0);
      if (k0 + NBk < Nn) {
        int mt32 = (Nn - k0 - NBk) >> 5;
        dim3 g(Mm / 32, (mt32 + 3) / 4);
        trsm_update_kernel<<<g, 128, 0, stream>>>(K, B, k0);
      }
    }

    // 6) predictive variance accumulation
    var_acc_kernel<<<Mm / 256, 256, 0, stream>>>(B, accV);
  }

  finalize_kernel<<<(Mm + 255) / 256, 256, 0, stream>>>(accM, accV, out);
}